// LSTMLM_Model_67645734913067
// MI455X (gfx1250) — compile-verified
//
#include <hip/hip_runtime.h>
#include <math.h>

// LSTM LM: V=32000, E=H=1024, L=2, B=32, T=128.
// fp32 WMMA (v_wmma_f32_16x16x4_f32) for all GEMMs.
// A tiles staged via LDS; B streams software-pipelined at depth 2 with the
// K-loop kept ROLLED (unroll(disable)) so register pressure stays bounded
// (no scratch spills) while each load group has ~8 WMMAs of latency cover.

#define V_ 32000
#define E_ 1024
#define H_ 1024
#define L_ 2
#define B_ 32
#define T_ 128

// LDS row pitch for a 16 x 1024 f32 A-tile: 1028 floats => consecutive rows
// are 4 banks apart; the 32-lane b64 fragment read covers all 64 banks once.
#define APITCH 1028
#define ATILE_BYTES (16 * APITCH * 4)   // 65792 B (dynamic LDS)

typedef __attribute__((ext_vector_type(2))) float v2f;
typedef __attribute__((ext_vector_type(8))) float v8f;

__device__ __forceinline__ v8f wmma4(v2f a, v2f b, v8f c) {
  // D = A(16x4) * B(4x16) + C(16x16), all f32.
  return __builtin_amdgcn_wmma_f32_16x16x4_f32(
      /*neg_a=*/false, a, /*neg_b=*/false, b,
      /*c_mod=*/(short)0, c, /*reuse_a=*/false, /*reuse_b=*/false);
}

__device__ __forceinline__ float sigm(float x) {
  return 1.0f / (1.0f + __expf(-x));
}

// Stage one 16x1024 f32 slab (rows m0..m0+15 of a [*,1024] matrix) into LDS
// with APITCH row pitch. 128 threads, coalesced float4 loads.
__device__ __forceinline__ void stage_tile(float* __restrict__ sA,
                                           const float* __restrict__ src,
                                           int m0) {
  const float4* g = (const float4*)(src + (size_t)m0 * H_);
#pragma unroll 4
  for (int i = threadIdx.x; i < 4096; i += 128) {
    const int row  = i >> 8;        // /256
    const int col4 = i & 255;
    float4 v = g[(size_t)row * 256 + col4];
    *(float4*)(sA + row * APITCH + col4 * 4) = v;
  }
}

// K-loop over one weight set (4 B-streams), software-pipelined, depth 2.
// Loop stays rolled: body = 10 loads + 8 WMMAs, two load groups in flight.
__device__ __forceinline__ void gemm_4stream(const float* __restrict__ sArow,
                                             const float* __restrict__ w0,
                                             const float* __restrict__ w1,
                                             const float* __restrict__ w2,
                                             const float* __restrict__ w3,
                                             v8f& acc0, v8f& acc1,
                                             v8f& acc2, v8f& acc3) {
  v2f a0  = *(const v2f*)(sArow + 0);
  v2f a1  = *(const v2f*)(sArow + 4);
  v2f c00 = *(const v2f*)(w0 + 0), c01 = *(const v2f*)(w1 + 0);
  v2f c02 = *(const v2f*)(w2 + 0), c03 = *(const v2f*)(w3 + 0);
  v2f c10 = *(const v2f*)(w0 + 4), c11 = *(const v2f*)(w1 + 4);
  v2f c12 = *(const v2f*)(w2 + 4), c13 = *(const v2f*)(w3 + 4);

#pragma clang loop unroll(disable)
  for (int k = 8; k < H_; k += 8) {
    acc0 = wmma4(a0, c00, acc0);
    acc1 = wmma4(a0, c01, acc1);
    acc2 = wmma4(a0, c02, acc2);
    acc3 = wmma4(a0, c03, acc3);
    a0  = *(const v2f*)(sArow + k);
    c00 = *(const v2f*)(w0 + k);
    c01 = *(const v2f*)(w1 + k);
    c02 = *(const v2f*)(w2 + k);
    c03 = *(const v2f*)(w3 + k);
    acc0 = wmma4(a1, c10, acc0);
    acc1 = wmma4(a1, c11, acc1);
    acc2 = wmma4(a1, c12, acc2);
    acc3 = wmma4(a1, c13, acc3);
    a1  = *(const v2f*)(sArow + k + 4);
    c10 = *(const v2f*)(w0 + k + 4);
    c11 = *(const v2f*)(w1 + k + 4);
    c12 = *(const v2f*)(w2 + k + 4);
    c13 = *(const v2f*)(w3 + k + 4);
  }
  // Drain both in-flight groups.
  acc0 = wmma4(a0, c00, acc0);
  acc1 = wmma4(a0, c01, acc1);
  acc2 = wmma4(a0, c02, acc2);
  acc3 = wmma4(a0, c03, acc3);
  acc0 = wmma4(a1, c10, acc0);
  acc1 = wmma4(a1, c11, acc1);
  acc2 = wmma4(a1, c12, acc2);
  acc3 = wmma4(a1, c13, acc3);
}

// ---------------------------------------------------------------------------
// Embedding gather: xs[t][b][:] = emb[tokens[b][t]][:]
// ---------------------------------------------------------------------------
__global__ void embed_kernel(const int* __restrict__ tokens,
                             const float* __restrict__ emb,
                             float* __restrict__ xs) {
  const int tb = blockIdx.x;      // tb = t*B + b  (xs is [T,B,E])
  const int t = tb / B_;
  const int b = tb - t * B_;
  const int tok = tokens[b * T_ + t];   // tokens is [B,T]
  const float4* src = (const float4*)(emb + (size_t)tok * E_);
  float4* dst = (float4*)(xs + (size_t)tb * E_);
  dst[threadIdx.x] = src[threadIdx.x];
}

// ---------------------------------------------------------------------------
// One LSTM cell step for one layer (gate order i,f,g,o):
//   gates = X @ Wih^T + Hprev @ Whh^T + bih + bhh
//   c_new = sig(f)*c + sig(i)*tanh(g);  h_new = sig(o)*tanh(c_new)
// Block = 128 threads = 4 waves sharing one batch M-tile; wave w owns htile
// hgrp*4+w and accumulates the 4 gate columns so the elementwise update is
// fused in-register. X and Hprev tiles staged through one LDS buffer in two
// phases. Grid = 2 mtiles * 16 hgroups = 32 blocks.
// ---------------------------------------------------------------------------
__global__ void lstm_step_kernel(const float* __restrict__ X,
                                 const float* __restrict__ Hprev,
                                 const float* __restrict__ Cprev,
                                 const float* __restrict__ Wih,
                                 const float* __restrict__ Whh,
                                 const float* __restrict__ bih,
                                 const float* __restrict__ bhh,
                                 float* __restrict__ Hnew,
                                 float* __restrict__ Cnew,
                                 float* __restrict__ TopH) {  // nullable
  extern __shared__ float sA[];     // 16 x APITCH floats

  const int wave  = threadIdx.x >> 5;
  const int lane  = threadIdx.x & 31;
  const int half  = lane >> 4;
  const int l16   = lane & 15;
  const int mtile = blockIdx.x >> 4;            // 0..1
  const int hgrp  = blockIdx.x & 15;            // 0..15
  const int htile = hgrp * 4 + wave;            // 0..63
  const int m0    = mtile * 16;
  const int hcol  = htile * 16 + l16;           // D column owned by this lane
  const int koff  = half * 2;                   // f32 A/B lane-half K offset

  const v8f zero = {0.f, 0.f, 0.f, 0.f, 0.f, 0.f, 0.f, 0.f};
  v8f acc_i = zero, acc_f = zero, acc_g = zero, acc_o = zero;

  const float* sArow = sA + l16 * APITCH + koff;     // A fragment base (LDS)

  // Phase 1: input contribution.
  stage_tile(sA, X, m0);
  __syncthreads();
  gemm_4stream(sArow,
               Wih + (size_t)(0 * H_ + hcol) * H_ + koff,
               Wih + (size_t)(1 * H_ + hcol) * H_ + koff,
               Wih + (size_t)(2 * H_ + hcol) * H_ + koff,
               Wih + (size_t)(3 * H_ + hcol) * H_ + koff,
               acc_i, acc_f, acc_g, acc_o);
  __syncthreads();

  // Phase 2: recurrent contribution.
  stage_tile(sA, Hprev, m0);
  __syncthreads();
  gemm_4stream(sArow,
               Whh + (size_t)(0 * H_ + hcol) * H_ + koff,
               Whh + (size_t)(1 * H_ + hcol) * H_ + koff,
               Whh + (size_t)(2 * H_ + hcol) * H_ + koff,
               Whh + (size_t)(3 * H_ + hcol) * H_ + koff,
               acc_i, acc_f, acc_g, acc_o);

  const float bi = bih[0 * H_ + hcol] + bhh[0 * H_ + hcol];
  const float bf = bih[1 * H_ + hcol] + bhh[1 * H_ + hcol];
  const float bg = bih[2 * H_ + hcol] + bhh[2 * H_ + hcol];
  const float bo = bih[3 * H_ + hcol] + bhh[3 * H_ + hcol];

#pragma unroll
  for (int r = 0; r < 8; ++r) {
    const int brow = m0 + half * 8 + r;          // D row M = 8*half + r
    const size_t idx = (size_t)brow * H_ + hcol;
    const float gi = sigm(acc_i[r] + bi);
    const float gf = sigm(acc_f[r] + bf);
    const float gg = tanhf(acc_g[r] + bg);
    const float go = sigm(acc_o[r] + bo);
    const float cn = gf * Cprev[idx] + gi * gg;
    const float hn = go * tanhf(cn);
    Cnew[idx] = cn;
    Hnew[idx] = hn;
    if (TopH) TopH[idx] = hn;
  }
}

// ---------------------------------------------------------------------------
// Output projection: Out[row][v] = top_h[row][:] . U[:][v] + b2
// M = T*B = 4096 rows, N = 32000, K = 1024.
// Block = 128 threads = 4 waves sharing one A tile (via LDS); each wave does
// 4 N-tiles. B loads (columns of U) are coalesced and software-pipelined.
// Grid = 256 mtiles * 125 ngroups = 32000 blocks. NT output stores.
// ---------------------------------------------------------------------------
__device__ __forceinline__ void ldB4(const float* __restrict__ U,
                                     int k, int koff, int colb,
                                     v2f& b0, v2f& b1, v2f& b2, v2f& b3) {
  const float* u0 = U + (size_t)(k + koff) * V_ + colb;  // row k+koff
  const float* u1 = u0 + V_;                             // row k+koff+1
  b0.x = u0[0];  b0.y = u1[0];
  b1.x = u0[16]; b1.y = u1[16];
  b2.x = u0[32]; b2.y = u1[32];
  b3.x = u0[48]; b3.y = u1[48];
}

__global__ void proj_kernel(const float* __restrict__ A,  // [4096,1024]
                            const float* __restrict__ U,  // [1024,32000]
                            const float* __restrict__ pb2,
                            float* __restrict__ Out) {    // [4096,32000]
  extern __shared__ float sA[];     // 16 x APITCH floats

  const int wave  = threadIdx.x >> 5;
  const int lane  = threadIdx.x & 31;
  const int half  = lane >> 4;
  const int l16   = lane & 15;
  const int nblk  = blockIdx.x % 125;           // 0..124
  const int mtile = blockIdx.x / 125;           // 0..255
  const int koff  = half * 2;
  const int colb  = nblk * 256 + wave * 64 + l16;

  stage_tile(sA, A, mtile * 16);
  __syncthreads();

  const float* sArow = sA + l16 * APITCH + koff;

  const v8f zero = {0.f, 0.f, 0.f, 0.f, 0.f, 0.f, 0.f, 0.f};
  v8f acc0 = zero, acc1 = zero, acc2 = zero, acc3 = zero;

  // Software pipeline, depth 2 (two k-step groups in flight), rolled loop.
  v2f a0 = *(const v2f*)(sArow + 0);
  v2f a1 = *(const v2f*)(sArow + 4);
  v2f c00, c01, c02, c03, c10, c11, c12, c13;
  ldB4(U, 0, koff, colb, c00, c01, c02, c03);
  ldB4(U, 4, koff, colb, c10, c11, c12, c13);

#pragma clang loop unroll(disable)
  for (int k = 8; k < H_; k += 8) {
    acc0 = wmma4(a0, c00, acc0);
    acc1 = wmma4(a0, c01, acc1);
    acc2 = wmma4(a0, c02, acc2);
    acc3 = wmma4(a0, c03, acc3);
    a0 = *(const v2f*)(sArow + k);
    ldB4(U, k, koff, colb, c00, c01, c02, c03);
    acc0 = wmma4(a1, c10, acc0);
    acc1 = wmma4(a1, c11, acc1);
    acc2 = wmma4(a1, c12, acc2);
    acc3 = wmma4(a1, c13, acc3);
    a1 = *(const v2f*)(sArow + k + 4);
    ldB4(U, k + 4, koff, colb, c10, c11, c12, c13);
  }
  acc0 = wmma4(a0, c00, acc0);
  acc1 = wmma4(a0, c01, acc1);
  acc2 = wmma4(a0, c02, acc2);
  acc3 = wmma4(a0, c03, acc3);
  acc0 = wmma4(a1, c10, acc0);
  acc1 = wmma4(a1, c11, acc1);
  acc2 = wmma4(a1, c12, acc2);
  acc3 = wmma4(a1, c13, acc3);

  const float bias = pb2[0];
#pragma unroll
  for (int r = 0; r < 8; ++r) {
    const size_t row = (size_t)(mtile * 16 + half * 8 + r);
    float* o = Out + row * V_ + colb;
    __builtin_nontemporal_store(acc0[r] + bias, o + 0);
    __builtin_nontemporal_store(acc1[r] + bias, o + 16);
    __builtin_nontemporal_store(acc2[r] + bias, o + 32);
    __builtin_nontemporal_store(acc3[r] + bias, o + 48);
  }
}

// ---------------------------------------------------------------------------
// Host-side launcher.
// ---------------------------------------------------------------------------
extern "C" void kernel_launch(void* const* d_in, const int* in_sizes, int n_in,
                              void* d_out, int out_size, void* d_ws, size_t ws_size,
                              hipStream_t stream) {
  (void)in_sizes; (void)n_in; (void)out_size; (void)ws_size;

  const int*   tokens = (const int*)d_in[0];
  const float* h0     = (const float*)d_in[1];
  const float* c0     = (const float*)d_in[2];
  const float* emb    = (const float*)d_in[3];
  const float* W_ih   = (const float*)d_in[4];
  const float* W_hh   = (const float*)d_in[5];
  const float* b_ih   = (const float*)d_in[6];
  const float* b_hh   = (const float*)d_in[7];
  const float* U      = (const float*)d_in[8];
  const float* b2     = (const float*)d_in[9];
  float* out = (float*)d_out;

  // Workspace layout (floats): xs | top_h | hbuf(x2) | cbuf(x2)
  float* ws   = (float*)d_ws;
  float* xs   = ws;                                  // T*B*E
  float* toph = xs + (size_t)T_ * B_ * E_;           // T*B*H
  float* hbuf = toph + (size_t)T_ * B_ * H_;         // 2*L*B*H (ping-pong)
  float* cbuf = hbuf + 2 * (size_t)L_ * B_ * H_;     // 2*L*B*H

  const size_t stateBytes = (size_t)L_ * B_ * H_ * sizeof(float);
  hipMemcpyAsync(hbuf, h0, stateBytes, hipMemcpyDeviceToDevice, stream);
  hipMemcpyAsync(cbuf, c0, stateBytes, hipMemcpyDeviceToDevice, stream);

  embed_kernel<<<T_ * B_, 256, 0, stream>>>(tokens, emb, xs);

  const size_t lbh = (size_t)L_ * B_ * H_;
  for (int t = 0; t < T_; ++t) {
    const int cur = t & 1;
    const int nxt = cur ^ 1;
    for (int l = 0; l < L_; ++l) {
      const float* X = (l == 0)
          ? (xs + (size_t)t * B_ * E_)
          : (hbuf + (size_t)nxt * lbh + 0 * (size_t)B_ * H_);  // layer0 h_new
      const float* Hp = hbuf + (size_t)cur * lbh + (size_t)l * B_ * H_;
      const float* Cp = cbuf + (size_t)cur * lbh + (size_t)l * B_ * H_;
      float* Hn = hbuf + (size_t)nxt * lbh + (size_t)l * B_ * H_;
      float* Cn = cbuf + (size_t)nxt * lbh + (size_t)l * B_ * H_;
      float* Td = (l == L_ - 1) ? (toph + (size_t)t * B_ * H_) : nullptr;
      lstm_step_kernel<<<32, 128, ATILE_BYTES, stream>>>(
          X, Hp, Cp,
          W_ih + (size_t)l * 4 * H_ * H_, W_hh + (size_t)l * 4 * H_ * H_,
          b_ih + (size_t)l * 4 * H_,      b_hh + (size_t)l * 4 * H_,
          Hn, Cn, Td);
    }
  }

  proj_kernel<<<256 * 125, 128, ATILE_BYTES, stream>>>(toph, U, b2, out);

  // Final states: last step (t=127) wrote parity 0.
  float* hfin = hbuf + 0;
  float* cfin = cbuf + 0;
  hipMemcpyAsync(out + (size_t)T_ * B_ * V_, hfin, stateBytes,
                 hipMemcpyDeviceToDevice, stream);
  hipMemcpyAsync(out + (size_t)T_ * B_ * V_ + lbh, cfin, stateBytes,
                 hipMemcpyDeviceToDevice, stream);
}